// QEmbedding_48859547959349
// MI455X (gfx1250) — compile-verified
//
#include <hip/hip_runtime.h>

typedef __attribute__((ext_vector_type(2))) float v2f;
typedef __attribute__((ext_vector_type(8))) float v8f;

#define QNUM 18
#define BSZ 256
#define NSTATE (1 << QNUM) /* 262144 */

// out[b,k] = prod_{q=0..17} ( bit_q(k) ? sin(t[b,17-q]/2) : cos(t[b,17-q]/2) )
//          = hi[h] * lo[m],   k = h*256 + m
// One block: batch row b, hi-slice of 256 h-values. 8 waves x 32 WMMA rank-1 tiles.
__global__ __launch_bounds__(256) void qembed_kron_wmma(
    const float* __restrict__ tokens, float* __restrict__ out)
{
    __shared__ float c_lds[QNUM];
    __shared__ float s_lds[QNUM];
    __shared__ float lo_lds[256];
    __shared__ float hi_lds[256];

    const int b     = blockIdx.x;   // batch row
    const int slice = blockIdx.y;   // h in [slice*256, slice*256+256)
    const int t     = threadIdx.x;

    // --- per-row angle factors ---
    if (t < QNUM) {
        float half = tokens[b * QNUM + t] * 0.5f;
        float sv, cv;
        sincosf(half, &sv, &cv);
        s_lds[t] = sv;
        c_lds[t] = cv;
    }
    __syncthreads();

    // --- lo table: bits q=0..7 of k select token i = 17-q ---
    {
        float p = 1.0f;
#pragma unroll
        for (int q = 0; q < 8; ++q) {
            int i = 17 - q;
            p *= ((t >> q) & 1) ? s_lds[i] : c_lds[i];
        }
        lo_lds[t] = p;
    }
    // --- hi table for this slice: bit j of h selects token i = 9-j ---
    {
        unsigned hg = (unsigned)slice * 256u + (unsigned)t;
        float p = 1.0f;
#pragma unroll
        for (int j = 0; j < 10; ++j) {
            int i = 9 - j;
            p *= ((hg >> j) & 1) ? s_lds[i] : c_lds[i];
        }
        hi_lds[t] = p;
    }
    __syncthreads();

    // --- rank-1 outer-product tiles via V_WMMA_F32_16X16X4_F32 ---
    const int lane   = t & 31;        // wave32
    const int wave   = t >> 5;        // 0..7
    const int lane15 = lane & 15;
    const int laneHi = lane >> 4;     // 0: lanes 0-15, 1: lanes 16-31

    // Bitmask to zero A in lanes 16-31 (K=2 slot) without exec-masked loads.
    const unsigned laneMask = (lane < 16) ? 0xFFFFFFFFu : 0u;

    float* const outBase = out + (size_t)b * NSTATE + (size_t)slice * (256 * 256);

#pragma unroll
    for (int hrep = 0; hrep < 2; ++hrep) {
        const int Ht = wave * 2 + hrep;         // hi-tile 0..15 within slice
        // A (16x4 f32): lanes 0-15 hold M=0..15 @ K=0 (VGPR0); K=1..3 forced 0.
        // Zeroing all of A's K=1..3 makes B's rows 1..3 don't-care.
        float av = hi_lds[Ht * 16 + lane15];    // unconditional, all lanes
        v2f A;
        A.x = __uint_as_float(__float_as_uint(av) & laneMask);
        A.y = 0.0f;
#pragma unroll
        for (int Mt = 0; Mt < 16; ++Mt) {       // lo-tile 0..15
            // B (4x16 f32): lanes 0-15 = row K=0 (used); lanes 16-31 = row K=2
            // (broadcast duplicate, multiplied by A[M,2]==0 -> harmless).
            v2f Bm;
            Bm.x = lo_lds[Mt * 16 + lane15];    // unconditional, all lanes
            Bm.y = 0.0f;                        // row K=1/K=3 = 0
            v8f Cz = {};
            // D[M,N] = hi[Ht*16+M] * lo[Mt*16+N]  (exact f32, RNE)
            v8f D = __builtin_amdgcn_wmma_f32_16x16x4_f32(
                false, A, false, Bm, (short)0, Cz, false, false);

            // D layout: VGPR v, lane l -> (M = v + 8*(l>=16), N = l%16)
            float* p = outBase
                     + (size_t)(Ht * 16 + laneHi * 8) * 256
                     + (size_t)(Mt * 16 + lane15);
#pragma unroll
            for (int v = 0; v < 8; ++v)
                __builtin_nontemporal_store(D[v], p + (size_t)v * 256);
        }
    }
}

extern "C" void kernel_launch(void* const* d_in, const int* in_sizes, int n_in,
                              void* d_out, int out_size, void* d_ws, size_t ws_size,
                              hipStream_t stream) {
    (void)in_sizes; (void)n_in; (void)d_ws; (void)ws_size; (void)out_size;
    const float* tokens = (const float*)d_in[0];   // [256, 18] fp32
    float* out = (float*)d_out;                    // [256, 262144] fp32
    dim3 grid(BSZ, 4, 1);                          // 1024 blocks, 8 waves each
    qembed_kron_wmma<<<grid, dim3(256, 1, 1), 0, stream>>>(tokens, out);
}